// mGNN_8589935117
// MI455X (gfx1250) — compile-verified
//
#include <hip/hip_runtime.h>

typedef __attribute__((ext_vector_type(2))) float v2f;
typedef __attribute__((ext_vector_type(8))) float v8f;

static __device__ __forceinline__ float lrelu(float v, float s) {
  return v > 0.0f ? v : v * s;
}

__global__ void fill_f32(float* __restrict__ p, float v, int n) {
  int i = blockIdx.x * blockDim.x + threadIdx.x;
  if (i < n) p[i] = v;
}

// C[M,Kout] = X[M,K] @ W[K,Kout], fp32 via V_WMMA_F32_16X16X4_F32.
// One wave computes one 16x16 C tile. Requires M%16==0, K%4==0, Kout%16==0
// so EXEC is all-ones everywhere (WMMA requirement), no predication needed.
// A fragment (16x4): lanes 0-15 hold rows M=0..15 with K,K+1 in v[0],v[1];
// lanes 16-31 hold K+2,K+3.  B fragment (4x16) mirrored across lanes.
__global__ void gemm_wmma_f32(const float* __restrict__ X,
                              const float* __restrict__ W,
                              float* __restrict__ C,
                              int K, int Kout, int tiles_n) {
  int wave = blockIdx.x * (blockDim.x >> 5) + (threadIdx.x >> 5);
  int tm = wave / tiles_n;
  int tn = wave - tm * tiles_n;
  int lane = threadIdx.x & 31;
  int rlo = lane & 15;
  int khalf = lane >> 4;  // 0 -> K,K+1 ; 1 -> K+2,K+3
  const float* xrow = X + (size_t)(tm * 16 + rlo) * K + khalf * 2;
  const float* wcol = W + (size_t)(khalf * 2) * Kout + tn * 16 + rlo;
  v8f acc = {};
  for (int k = 0; k < K; k += 4) {
    v2f a, b;
    a.x = xrow[k];
    a.y = xrow[k + 1];
    b.x = wcol[(size_t)k * Kout];
    b.y = wcol[(size_t)(k + 1) * Kout];
    acc = __builtin_amdgcn_wmma_f32_16x16x4_f32(false, a, false, b, (short)0,
                                                acc, false, false);
  }
  // C/D layout: lanes 0-15 -> rows M=0..7 in v0..v7 (col = lane),
  //             lanes 16-31 -> rows M=8..15 (col = lane-16).
  float* crow = C + (size_t)(tm * 16 + khalf * 8) * Kout + tn * 16 + rlo;
#pragma unroll
  for (int i = 0; i < 8; i++) crow[(size_t)i * Kout] = acc[i];
}

// el[n,h] = dot(feat[n,h,:], al[h,:]);  er likewise.
__global__ void attn_coef(const float* __restrict__ feat,
                          const float* __restrict__ al,
                          const float* __restrict__ ar,
                          float* __restrict__ el, float* __restrict__ er,
                          int n, int H, int F) {
  int i = blockIdx.x * blockDim.x + threadIdx.x;
  if (i >= n * H) return;
  int node = i / H, h = i - node * H;
  const float* fp = feat + (size_t)node * H * F + (size_t)h * F;
  const float* alp = al + (size_t)h * F;
  const float* arp = ar + (size_t)h * F;
  float sl = 0.0f, sr = 0.0f;
  for (int f = 0; f < F; ++f) {
    float v = fp[f];
    sl += v * alp[f];
    sr += v * arp[f];
  }
  el[i] = sl;
  er[i] = sr;
}

static __device__ __forceinline__ void atomicMaxF(float* addr, float val) {
  int* ia = (int*)addr;
  int old = __float_as_int(*addr);
  while (__int_as_float(old) < val) {
    int assumed = old;
    old = atomicCAS(ia, assumed, __float_as_int(val));
    if (old == assumed) break;
  }
}

__global__ void edge_max(const int* __restrict__ src, const int* __restrict__ dst,
                         const float* __restrict__ el, const float* __restrict__ er,
                         float* __restrict__ emax, int E, int H) {
  int i = blockIdx.x * blockDim.x + threadIdx.x;
  if (i >= E * H) return;
  int e = i / H, h = i - e * H;
  int s = src[e], d = dst[e];
  float v = lrelu(el[s * H + h] + er[d * H + h], 0.2f);
  atomicMaxF(&emax[d * H + h], v);
}

__global__ void edge_expsum(const int* __restrict__ src, const int* __restrict__ dst,
                            const float* __restrict__ el, const float* __restrict__ er,
                            const float* __restrict__ emax, float* __restrict__ ee,
                            float* __restrict__ denom, int E, int H) {
  int i = blockIdx.x * blockDim.x + threadIdx.x;
  if (i >= E * H) return;
  int e = i / H, h = i - e * H;
  int s = src[e], d = dst[e];
  float v = lrelu(el[s * H + h] + er[d * H + h], 0.2f);
  float x = expf(v - emax[d * H + h]);
  ee[i] = x;
  atomicAdd(&denom[d * H + h], x);
}

// rst[dst, coloff + (h,f)] += feat[src,h,f] * ee[e,h]/denom[dst,h]
// tid -> (edge, flat h*F+f): consecutive lanes touch consecutive addresses,
// so both the feat gather (L2-resident) and the atomic scatter are coalesced.
__global__ void edge_scatter(const int* __restrict__ src, const int* __restrict__ dst,
                             const float* __restrict__ feat,
                             const float* __restrict__ ee,
                             const float* __restrict__ denom,
                             float* __restrict__ rst,
                             int E, int H, int F, int rstride, int coloff) {
  long long i = (long long)blockIdx.x * blockDim.x + threadIdx.x;
  int hf = H * F;
  long long total = (long long)E * hf;
  if (i >= total) return;
  int e = (int)(i / hf);
  int r = (int)(i - (long long)e * hf);
  int h = r / F;
  int s = src[e], d = dst[e];
  float a = ee[e * H + h] / denom[d * H + h];
  atomicAdd(&rst[(size_t)d * rstride + coloff + r],
            feat[(size_t)s * hf + r] * a);
}

// h = leaky_relu(concat(rst_c1 + b_c1, rst_a1 + b_a1), 0.01)
__global__ void make_h(const float* __restrict__ rc, const float* __restrict__ ra,
                       const float* __restrict__ bc, const float* __restrict__ ba,
                       float* __restrict__ h, int n, int hw /*=H*F1*/) {
  int i = blockIdx.x * blockDim.x + threadIdx.x;
  if (i >= n * 2 * hw) return;
  int node = i / (2 * hw);
  int j = i - node * 2 * hw;
  float v = (j < hw) ? (rc[(size_t)node * hw + j] + bc[j])
                     : (ra[(size_t)node * hw + (j - hw)] + ba[j - hw]);
  h[i] = lrelu(v, 0.01f);
}

__global__ void add_bias_out(float* __restrict__ out, const float* __restrict__ bc,
                             const float* __restrict__ ba, int n, int hw /*=H*F2*/) {
  int i = blockIdx.x * blockDim.x + threadIdx.x;
  if (i >= n * 2 * hw) return;
  int j = i % (2 * hw);
  out[i] += (j < hw) ? bc[j] : ba[j - hw];
}

extern "C" void kernel_launch(void* const* d_in, const int* in_sizes, int n_in,
                              void* d_out, int out_size, void* d_ws, size_t ws_size,
                              hipStream_t stream) {
  const float* x    = (const float*)d_in[0];
  const int* src1   = (const int*)d_in[1];
  const int* dst1   = (const int*)d_in[2];
  const int* src2   = (const int*)d_in[3];
  const int* dst2   = (const int*)d_in[4];
  const float* W_c1 = (const float*)d_in[5];
  const float* al_c1 = (const float*)d_in[6];
  const float* ar_c1 = (const float*)d_in[7];
  const float* b_c1 = (const float*)d_in[8];
  const float* W_a1 = (const float*)d_in[9];
  const float* al_a1 = (const float*)d_in[10];
  const float* ar_a1 = (const float*)d_in[11];
  const float* b_a1 = (const float*)d_in[12];
  const float* W_c2 = (const float*)d_in[13];
  const float* al_c2 = (const float*)d_in[14];
  const float* ar_c2 = (const float*)d_in[15];
  const float* b_c2 = (const float*)d_in[16];
  const float* W_a2 = (const float*)d_in[17];
  const float* al_a2 = (const float*)d_in[18];
  const float* ar_a2 = (const float*)d_in[19];
  const float* b_a2 = (const float*)d_in[20];

  const int INF_ = 128, H = 4, F1 = 128, F2 = 64;
  const int N = in_sizes[0] / INF_;
  const int E = in_sizes[1];
  const int HF1 = H * F1;      // 512
  const int HF2 = H * F2;      // 256
  const int D2 = 2 * HF1;      // 1024

  // Workspace carving (floats). Layer-2 feat buffers reuse layer-1's.
  float* ws = (float*)d_ws;
  float* feat_c = ws;                               // N*HF1 (layer2 uses N*HF2)
  float* feat_a = feat_c + (size_t)N * HF1;         // N*HF1
  float* rst_c1 = feat_a + (size_t)N * HF1;         // N*HF1
  float* rst_a1 = rst_c1 + (size_t)N * HF1;         // N*HF1
  float* hbuf   = rst_a1 + (size_t)N * HF1;         // N*D2
  float* ee     = hbuf + (size_t)N * D2;            // E*H
  float* el     = ee + (size_t)E * H;               // N*H
  float* er     = el + (size_t)N * H;               // N*H
  float* emax   = er + (size_t)N * H;               // N*H
  float* denom  = emax + (size_t)N * H;             // N*H

  float* out = (float*)d_out;
  const int T = 256;
  auto blocks = [](long long n) { return (int)((n + 255) / 256); };

  // Zero the layer-2 accumulator (d_out) up front (harness poisons it).
  fill_f32<<<blocks((long long)N * 2 * HF2), T, 0, stream>>>(out, 0.0f, N * 2 * HF2);

  auto run_conv = [&](const float* xin, int Kin, const int* src, const int* dst,
                      const float* W, const float* al, const float* ar,
                      float* feat, int F, float* rst, int rstride, int coloff,
                      bool zero_rst) {
    int HF = H * F;
    int tiles = (N / 16) * (HF / 16);                 // divisible by 8 waves/block
    gemm_wmma_f32<<<tiles / 8, 256, 0, stream>>>(xin, W, feat, Kin, HF, HF / 16);
    attn_coef<<<blocks((long long)N * H), T, 0, stream>>>(feat, al, ar, el, er, N, H, F);
    fill_f32<<<blocks((long long)N * H), T, 0, stream>>>(emax, -3.402823466e38f, N * H);
    fill_f32<<<blocks((long long)N * H), T, 0, stream>>>(denom, 0.0f, N * H);
    if (zero_rst)
      fill_f32<<<blocks((long long)N * HF), T, 0, stream>>>(rst, 0.0f, N * HF);
    edge_max<<<blocks((long long)E * H), T, 0, stream>>>(src, dst, el, er, emax, E, H);
    edge_expsum<<<blocks((long long)E * H), T, 0, stream>>>(src, dst, el, er, emax, ee,
                                                            denom, E, H);
    edge_scatter<<<blocks((long long)E * HF), T, 0, stream>>>(src, dst, feat, ee, denom,
                                                              rst, E, H, F, rstride,
                                                              coloff);
  };

  // Layer 1: two convs over the two edge sets.
  run_conv(x, INF_, src1, dst1, W_c1, al_c1, ar_c1, feat_c, F1, rst_c1, HF1, 0, true);
  run_conv(x, INF_, src2, dst2, W_a1, al_a1, ar_a1, feat_a, F1, rst_a1, HF1, 0, true);

  // h = leaky_relu(concat(rst_c1 + b_c1, rst_a1 + b_a1), 0.01)
  make_h<<<blocks((long long)N * D2), T, 0, stream>>>(rst_c1, rst_a1, b_c1, b_a1,
                                                      hbuf, N, HF1);

  // Layer 2: scatter straight into d_out (row stride 2*HF2, column offsets 0 / HF2).
  run_conv(hbuf, D2, src1, dst1, W_c2, al_c2, ar_c2, feat_c, F2, out, 2 * HF2, 0, false);
  run_conv(hbuf, D2, src2, dst2, W_a2, al_a2, ar_a2, feat_a, F2, out, 2 * HF2, HF2, false);

  // Final biases for the concatenated layer-2 output.
  add_bias_out<<<blocks((long long)N * 2 * HF2), T, 0, stream>>>(out, b_c2, b_a2, N, HF2);

  (void)n_in; (void)out_size; (void)ws_size;
}